// RandomHorizontalFlip_41145786696321
// MI455X (gfx1250) — compile-verified
//
#include <hip/hip_runtime.h>
#include <hip/hip_bf16.h>

// RandomHorizontalFlip: out[b,c,h,w] = flip[b] ? in[b,c,h,223-w] : in[b,c,h,w]
// Pure streaming permutation, ~308 MB total traffic -> HBM-bandwidth bound
// (~13 us floor @ 23.3 TB/s). Strategy: 128-bit NT loads/stores, wave-uniform
// flip flag (scalar branch), both read and write fully coalesced even on the
// reversed path (lanes walk descending-but-contiguous addresses).

typedef float v4f __attribute__((ext_vector_type(4)));

// Fixed problem geometry from the reference: W = 224 -> 56 float4 per row.
#define W4      56      // float4s per row
#define W4_LAST 55      // W4 - 1
#define PERB4   37632   // 3 * 224 * 56 float4s per batch sample
#define TPB     256     // 8 wave32 per block

__global__ __launch_bounds__(TPB) void hflip_kernel(
    const v4f* __restrict__ in,     // [B * PERB4] float4
    const int* __restrict__ flip,   // [B]
    v4f*       __restrict__ out)    // [B * PERB4] float4
{
    const int b   = blockIdx.x;                       // batch sample (uniform)
    const int tid = blockIdx.y * TPB + threadIdx.x;   // 0 .. PERB4-1 (exact)
    const long base = (long)b * PERB4;
    const long dst  = base + tid;                     // same dst both paths

    // flip[b] is wave-uniform -> s_load + scalar branch, no divergence.
    if (flip[b] != 0) {
        const int w4 = tid % W4;          // position within row (float4 units)
        const int cr = tid / W4;          // (c*H + h) row index
        const long src = base + (long)cr * W4 + (W4_LAST - w4);
        v4f v = __builtin_nontemporal_load(&in[src]);
        v4f r;
        r.x = v.w; r.y = v.z; r.z = v.y; r.w = v.x;   // reverse inside float4
        __builtin_nontemporal_store(r, &out[dst]);
    } else {
        v4f v = __builtin_nontemporal_load(&in[dst]);
        __builtin_nontemporal_store(v, &out[dst]);
    }
}

extern "C" void kernel_launch(void* const* d_in, const int* in_sizes, int n_in,
                              void* d_out, int out_size, void* d_ws, size_t ws_size,
                              hipStream_t stream)
{
    const v4f* x    = (const v4f*)d_in[0];   // [256,3,224,224] fp32, 16B-aligned rows
    const int* flip = (const int*)d_in[1];   // [256] int32
    v4f*       out  = (v4f*)d_out;

    const int B = in_sizes[1];               // 256
    // PERB4 = 37632 = 147 * 256 exactly -> no bounds check needed.
    dim3 grid((unsigned)B, PERB4 / TPB);     // (256, 147)
    hflip_kernel<<<grid, TPB, 0, stream>>>(x, flip, out);
}